// ContrastiveLossWithinBatch_84688165142763
// MI455X (gfx1250) — compile-verified
//
#include <hip/hip_runtime.h>
#include <math.h>

// ---------------------------------------------------------------------------
// Triplet/contrastive loss within batch, fused on CDNA5 (gfx1250, wave32).
//   x: [4096,1024] f32, label: [4096] i32  ->  scalar f32 loss
// Gram matrix via V_WMMA_F32_16X16X4_F32 on raw x, rescaled by per-row inverse
// norms; per-row reductions fused (never materialize the 4096x4096 matrix).
// ---------------------------------------------------------------------------

typedef __attribute__((ext_vector_type(2))) float v2f;
typedef __attribute__((ext_vector_type(8))) float v8f;

#define BATCH 4096
#define CDIM  1024
#define NIB   (BATCH / 16)   // 256 row-blocks
#define MARGIN 0.5f
#define EPSN   1e-12f
#define INF_BITS 0x7F800000u

// ---------------------------------------------------------------------------
// Kernel 1: per-row stats.  inv[r] = 1/max(||x_r||, eps);  sq[r] = ss*inv^2
// ---------------------------------------------------------------------------
__global__ __launch_bounds__(256) void row_stats_kernel(
    const float* __restrict__ x, float* __restrict__ inv, float* __restrict__ sq)
{
    const int r   = blockIdx.x;
    const int tid = threadIdx.x;
    const float* row = x + (size_t)r * CDIM;

    float ss = 0.0f;
    for (int i = tid; i < CDIM; i += 256) {
        float v = row[i];
        ss += v * v;
    }
    // wave32 reduction
    for (int m = 16; m >= 1; m >>= 1) ss += __shfl_xor(ss, m, 32);

    __shared__ float s_w[8];
    const int wave = tid >> 5, lane = tid & 31;
    if (lane == 0) s_w[wave] = ss;
    __syncthreads();
    if (tid == 0) {
        float t = 0.0f;
        #pragma unroll
        for (int i = 0; i < 8; ++i) t += s_w[i];
        float nrm = sqrtf(t);
        float den = fmaxf(nrm, EPSN);
        float iv  = 1.0f / den;
        inv[r] = iv;
        sq[r]  = t * iv * iv;
    }
}

// ---------------------------------------------------------------------------
// Kernel 2: fused Gram + triplet reductions.
// One workgroup per 16-row block; 8 waves sweep all 256 column tiles.
// ---------------------------------------------------------------------------
__global__ __launch_bounds__(256) void gram_triplet_kernel(
    const float* __restrict__ x, const float* __restrict__ inv,
    const float* __restrict__ sq, const int* __restrict__ label,
    float* __restrict__ partials)
{
    const int ib   = blockIdx.x;          // 0..255 row block
    const int tid  = threadIdx.x;
    const int wave = tid >> 5;            // 0..7
    const int lane = tid & 31;
    const int half = lane >> 4;           // 0/1: C-layout M-half and A K-half
    const int l16  = lane & 15;

    __shared__ int      s_labi[16];
    __shared__ float    s_sqi[16];
    __shared__ float    s_invi[16];
    __shared__ float    s_pos[16];
    __shared__ unsigned s_max[16];
    __shared__ unsigned s_minneg[16];
    __shared__ unsigned s_minall[16];

    if (tid < 16) {
        const int r = ib * 16 + tid;
        s_labi[tid]   = label[r];
        s_sqi[tid]    = sq[r];
        s_invi[tid]   = inv[r];
        s_pos[tid]    = 0.0f;
        s_max[tid]    = 0u;          // distances are >= 0: uint order == float order
        s_minneg[tid] = INF_BITS;
        s_minall[tid] = INF_BITS;
    }
    __syncthreads();

    // Register copies of row-side quantities for the 8 rows this lane touches
    // (C/D layout: VGPR v holds M = v + 8*half for this lane).
    int   labR[8];
    float sqR[8], invR[8];
    #pragma unroll
    for (int v = 0; v < 8; ++v) {
        const int m = v + 8 * half;
        labR[v] = s_labi[m];
        sqR[v]  = s_sqi[m];
        invR[v] = s_invi[m];
    }

    // Per-lane running reductions per row (index v).
    float accPos[8], accMax[8], accMinNeg[8], accMinAll[8];
    const float FINF = __uint_as_float(INF_BITS);
    #pragma unroll
    for (int v = 0; v < 8; ++v) {
        accPos[v] = 0.0f; accMax[v] = 0.0f;
        accMinNeg[v] = FINF; accMinAll[v] = FINF;
    }

    // A fragment pointer: A[M][K] f32 16x4 layout ->
    //   lane (half,l16): row = ib*16 + l16, cols kc + 2*half + {0,1}
    const float* aptr = x + (size_t)(ib * 16 + l16) * CDIM + 2 * half;

    for (int jo = 0; jo < 8; ++jo) {
        const int jb0 = jo * 32 + wave * 4;   // 4 consecutive column tiles
        const float* bptr = x + (size_t)(jb0 * 16 + l16) * CDIM + 2 * half;

        v8f c0 = {}, c1 = {}, c2 = {}, c3 = {};
        #pragma unroll 4
        for (int kc = 0; kc < CDIM; kc += 4) {
            v2f a  = *(const v2f*)(aptr + kc);
            v2f b0 = *(const v2f*)(bptr + kc);
            v2f b1 = *(const v2f*)(bptr + (size_t)1 * 16 * CDIM + kc);
            v2f b2 = *(const v2f*)(bptr + (size_t)2 * 16 * CDIM + kc);
            v2f b3 = *(const v2f*)(bptr + (size_t)3 * 16 * CDIM + kc);
            c0 = __builtin_amdgcn_wmma_f32_16x16x4_f32(false, a, false, b0, (short)0, c0, false, false);
            c1 = __builtin_amdgcn_wmma_f32_16x16x4_f32(false, a, false, b1, (short)0, c1, false, false);
            c2 = __builtin_amdgcn_wmma_f32_16x16x4_f32(false, a, false, b2, (short)0, c2, false, false);
            c3 = __builtin_amdgcn_wmma_f32_16x16x4_f32(false, a, false, b3, (short)0, c3, false, false);
        }

        // Epilogue for the 4 tiles: distances + masked reductions.
        #pragma unroll
        for (int t = 0; t < 4; ++t) {
            const v8f c = (t == 0) ? c0 : (t == 1) ? c1 : (t == 2) ? c2 : c3;
            const int jb   = jb0 + t;
            const int colg = jb * 16 + l16;   // this lane's column (N = lane%16)
            const int   labj = label[colg];
            const float sqj  = sq[colg];
            const float invj = inv[colg];
            #pragma unroll
            for (int v = 0; v < 8; ++v) {
                const int m = v + 8 * half;
                const float dotn = c[v] * invR[v] * invj;     // normalized dot
                const float d = fmaxf(sqR[v] + sqj - 2.0f * dotn, 0.0f);
                const bool same = (labR[v] == labj);
                const bool diag = ((ib * 16 + m) == colg);
                if (same && !diag) accPos[v] += d;
                accMax[v]    = fmaxf(accMax[v], d);
                accMinAll[v] = fminf(accMinAll[v], d);
                if (!same)   accMinNeg[v] = fminf(accMinNeg[v], d);
            }
        }
    }

    // Cross-lane reduce within each 16-lane half (xor masks stay in-half).
    #pragma unroll
    for (int v = 0; v < 8; ++v) {
        for (int m = 1; m <= 8; m <<= 1) {
            accPos[v]    += __shfl_xor(accPos[v], m, 32);
            accMax[v]     = fmaxf(accMax[v],    __shfl_xor(accMax[v],    m, 32));
            accMinNeg[v]  = fminf(accMinNeg[v], __shfl_xor(accMinNeg[v], m, 32));
            accMinAll[v]  = fminf(accMinAll[v], __shfl_xor(accMinAll[v], m, 32));
        }
    }
    // Lane 0 of each half merges into LDS across the 8 waves.
    if (l16 == 0) {
        #pragma unroll
        for (int v = 0; v < 8; ++v) {
            const int m = v + 8 * half;
            atomicAdd(&s_pos[m], accPos[v]);
            atomicMax(&s_max[m],    __float_as_uint(accMax[v]));
            atomicMin(&s_minneg[m], __float_as_uint(accMinNeg[v]));
            atomicMin(&s_minall[m], __float_as_uint(accMinAll[v]));
        }
    }
    __syncthreads();

    // Final per-row loss + deterministic in-block sum (wave 0 only).
    if (tid < 32) {
        float loss = 0.0f;
        if (tid < 16) {
            const float pos = s_pos[tid];
            const float mx  = __uint_as_float(s_max[tid]);
            const unsigned mnb = s_minneg[tid];
            // hardest = min over negatives if any; else min_all + max (exact
            // emulation of d + max*(1-negmask) row-min).
            const float hardest = (mnb != INF_BITS)
                                    ? __uint_as_float(mnb)
                                    : (__uint_as_float(s_minall[tid]) + mx);
            loss = fmaxf(pos + MARGIN - hardest, 0.0f);
        }
        for (int m = 1; m <= 16; m <<= 1) loss += __shfl_xor(loss, m, 32);
        if (tid == 0) partials[ib] = loss * (1.0f / (float)BATCH);
    }
}

// ---------------------------------------------------------------------------
// Kernel 3: deterministic final reduction of 256 block partials.
// ---------------------------------------------------------------------------
__global__ __launch_bounds__(256) void final_reduce_kernel(
    const float* __restrict__ partials, float* __restrict__ out)
{
    const int tid = threadIdx.x;
    float v = partials[tid];
    for (int m = 16; m >= 1; m >>= 1) v += __shfl_xor(v, m, 32);
    __shared__ float s_w[8];
    if ((tid & 31) == 0) s_w[tid >> 5] = v;
    __syncthreads();
    if (tid == 0) {
        float t = 0.0f;
        #pragma unroll
        for (int i = 0; i < 8; ++i) t += s_w[i];
        out[0] = t;  // SCALE == 1.0
    }
}

// ---------------------------------------------------------------------------
extern "C" void kernel_launch(void* const* d_in, const int* in_sizes, int n_in,
                              void* d_out, int out_size, void* d_ws, size_t ws_size,
                              hipStream_t stream) {
    (void)in_sizes; (void)n_in; (void)out_size; (void)ws_size;
    const float* x     = (const float*)d_in[0];
    const int*   label = (const int*)d_in[1];
    float* out = (float*)d_out;

    // Workspace: inv[4096] | sq[4096] | partials[256]  (~33 KB)
    float* inv      = (float*)d_ws;
    float* sq       = inv + BATCH;
    float* partials = sq + BATCH;

    row_stats_kernel<<<BATCH, 256, 0, stream>>>(x, inv, sq);
    gram_triplet_kernel<<<NIB, 256, 0, stream>>>(x, inv, sq, label, partials);
    final_reduce_kernel<<<1, 256, 0, stream>>>(partials, out);
}